// DWT_32693291058016
// MI455X (gfx1250) — compile-verified
//
#include <hip/hip_runtime.h>

// ---------------------------------------------------------------------------
// 2-D Haar DWT, x: (8,512,512,32) f32 NHWC -> (ll,lh,hl,hh) each (8,256,256,32)
// Memory-bound: 256MB in + 256MB out, ~22us floor at 23.3 TB/s.
// Strategy: async global->LDS double-buffered pipeline (CDNA5 ASYNCcnt path),
// column-parity de-interleave in LDS, float4 coalesced stores.
// ---------------------------------------------------------------------------

#define N_   8
#define H_   512
#define W_   512
#define CH_  32
#define OH_  256
#define OW_  256
#define HS_  (W_*CH_)         // 16384 floats: input row stride
#define NS_  (H_*W_*CH_)      // 8388608 floats: input image stride
#define OIS_ (OW_*CH_)        // 8192 floats: output row stride
#define ONS_ (OH_*OW_*CH_)    // 2097152 floats: output image stride
#define OSZ_ ((size_t)N_*ONS_)// 16777216 floats per subband

#define C_HAAR 0.70710678118654752440f

#define AS1 __attribute__((address_space(1)))
#define AS3 __attribute__((address_space(3)))

typedef int v4i_t __attribute__((ext_vector_type(4)));
typedef AS1 v4i_t* g128_t;   // global 16B vector pointer (AS1)
typedef AS3 v4i_t* l128_t;   // LDS 16B vector pointer (AS3)

#if defined(__has_builtin)
#  if __has_builtin(__builtin_amdgcn_global_load_async_to_lds_b128)
#    define HAVE_ASYNC_LDS 1
#  else
#    define HAVE_ASYNC_LDS 0
#  endif
#  if __has_builtin(__builtin_amdgcn_s_wait_asynccnt)
#    define HAVE_WAIT_ASYNC 1
#  else
#    define HAVE_WAIT_ASYNC 0
#  endif
#else
#  define HAVE_ASYNC_LDS 0
#  define HAVE_WAIT_ASYNC 0
#endif

#if HAVE_WAIT_ASYNC
#  define WAIT_ASYNC(n) __builtin_amdgcn_s_wait_asynccnt(n)
#else
#  define WAIT_ASYNC(n) asm volatile("s_wait_asynccnt %0" :: "i"(n) : "memory")
#endif

__device__ __forceinline__ float4 add4(float4 a, float4 b) {
    return make_float4(a.x + b.x, a.y + b.y, a.z + b.z, a.w + b.w);
}
__device__ __forceinline__ float4 sub4(float4 a, float4 b) {
    return make_float4(a.x - b.x, a.y - b.y, a.z - b.z, a.w - b.w);
}
__device__ __forceinline__ float4 scale4(float4 a, float s) {
    return make_float4(a.x * s, a.y * s, a.z * s, a.w * s);
}

__device__ __forceinline__ void haar_butterfly(float4 x00, float4 x01,
                                               float4 x10, float4 x11,
                                               float4& ll, float4& lh,
                                               float4& hl, float4& hh) {
    // Two-stage (row then column) to mirror the reference's rounding order.
    float4 a0 = scale4(add4(x00, x01), C_HAAR);
    float4 d0 = scale4(sub4(x00, x01), C_HAAR);
    float4 a1 = scale4(add4(x10, x11), C_HAAR);
    float4 d1 = scale4(sub4(x10, x11), C_HAAR);
    ll = scale4(add4(a0, a1), C_HAAR);
    lh = scale4(sub4(a0, a1), C_HAAR);
    hl = scale4(add4(d0, d1), C_HAAR);
    hh = scale4(sub4(d0, d1), C_HAAR);
}

// ---------------------------------------------------------------------------
// Async-pipelined kernel: 1 block = 1 output row (n,i); 8 W-chunks of
// 64 input cols, double-buffered through LDS via GLOBAL_LOAD_ASYNC_TO_LDS_B128.
// LDS tile layout (per row r in {0,1}): even cols [0..255] f4, odd cols
// [256..511] f4 -> conflict-free ds reads in the compute phase.
// ---------------------------------------------------------------------------
__global__ __launch_bounds__(256) void haar_async(const float* __restrict__ x,
                                                  float* __restrict__ out) {
    __shared__ float4 lds[2][1024];  // 2 x 16KB double buffer

    const int tid = threadIdx.x;
    const int b   = blockIdx.x;
    const int i   = b & (OH_ - 1);
    const int n   = b >> 8;
    const size_t in_base = (size_t)n * NS_ + (size_t)(2 * i) * HS_;

    const int c4 = tid & 7;   // float4 group of channels
    const int jl = tid >> 3;  // local output column 0..31

    auto issue_tile = [&](int buf, int chunk) {
        const size_t gbase = in_base + (size_t)chunk * 64 * CH_;
#pragma unroll
        for (int k = 0; k < 4; ++k) {
            const int s   = tid + k * 256;       // float4 slot 0..1023
            const int row = s >> 9;              // input row 0/1 of the pair
            const int sr  = s & 511;
            const int w   = sr >> 3;             // input col 0..63 in chunk
            const int cg  = sr & 7;              // channel float4 group
            const float* gp = x + gbase + (size_t)row * HS_ + w * CH_ + cg * 4;
            // de-interleave by column parity while landing in LDS
            float4* lp = &lds[buf][row * 512 + ((w >> 1) + (w & 1) * 32) * 8 + cg];
#if HAVE_ASYNC_LDS
            __builtin_amdgcn_global_load_async_to_lds_b128(
                (g128_t)(float*)gp, (l128_t)lp, 0, 0);
#else
            *lp = *(const float4*)gp;
#endif
        }
    };

    issue_tile(0, 0);
#pragma unroll 2
    for (int chunk = 0; chunk < 8; ++chunk) {
        const int buf = chunk & 1;
        if (chunk + 1 < 8) {
            issue_tile(buf ^ 1, chunk + 1);
            WAIT_ASYNC(4);   // current tile's 4 copies done; next 4 in flight
        } else {
            WAIT_ASYNC(0);
        }
        __syncthreads();

        const float4 x00 = lds[buf][      jl * 8 + c4];  // row0 even col
        const float4 x01 = lds[buf][256 + jl * 8 + c4];  // row0 odd  col
        const float4 x10 = lds[buf][512 + jl * 8 + c4];  // row1 even col
        const float4 x11 = lds[buf][768 + jl * 8 + c4];  // row1 odd  col

        float4 ll, lh, hl, hh;
        haar_butterfly(x00, x01, x10, x11, ll, lh, hl, hh);

        const int j = chunk * 32 + jl;
        const size_t o = (size_t)n * ONS_ + (size_t)i * OIS_ + (size_t)j * CH_ + c4 * 4;
        *(float4*)(out + o)            = ll;
        *(float4*)(out + OSZ_ + o)     = lh;
        *(float4*)(out + 2 * OSZ_ + o) = hl;
        *(float4*)(out + 3 * OSZ_ + o) = hh;

        __syncthreads();  // all reads of buf done before it is overwritten
    }
}

// ---------------------------------------------------------------------------
// Fallback: direct fully-coalesced float4 streaming kernel.
// ---------------------------------------------------------------------------
__global__ __launch_bounds__(256) void haar_direct(const float* __restrict__ x,
                                                   float* __restrict__ out) {
    const size_t t = (size_t)blockIdx.x * blockDim.x + threadIdx.x;
    const int c4 = (int)(t & 7);
    const size_t p = t >> 3;
    const int j = (int)(p & (OW_ - 1));
    const int i = (int)((p >> 8) & (OH_ - 1));
    const int n = (int)(p >> 16);

    const float* bp = x + (size_t)n * NS_ + (size_t)(2 * i) * HS_ +
                      (size_t)(2 * j) * CH_ + c4 * 4;
    __builtin_prefetch(bp + 2 * HS_, 0, 0);  // global_prefetch_b8 for next row pair

    const float4 x00 = *(const float4*)(bp);
    const float4 x01 = *(const float4*)(bp + CH_);
    const float4 x10 = *(const float4*)(bp + HS_);
    const float4 x11 = *(const float4*)(bp + HS_ + CH_);

    float4 ll, lh, hl, hh;
    haar_butterfly(x00, x01, x10, x11, ll, lh, hl, hh);

    const size_t o = (size_t)n * ONS_ + (size_t)i * OIS_ + (size_t)j * CH_ + c4 * 4;
    *(float4*)(out + o)            = ll;
    *(float4*)(out + OSZ_ + o)     = lh;
    *(float4*)(out + 2 * OSZ_ + o) = hl;
    *(float4*)(out + 3 * OSZ_ + o) = hh;
}

extern "C" void kernel_launch(void* const* d_in, const int* in_sizes, int n_in,
                              void* d_out, int out_size, void* d_ws, size_t ws_size,
                              hipStream_t stream) {
    (void)in_sizes; (void)n_in; (void)out_size; (void)d_ws; (void)ws_size;
    const float* x = (const float*)d_in[0];
    float* out = (float*)d_out;
#if HAVE_ASYNC_LDS
    haar_async<<<N_ * OH_, 256, 0, stream>>>(x, out);
#else
    haar_direct<<<(N_ * OH_ * OW_ * 8) / 256, 256, 0, stream>>>(x, out);
#endif
}